// MHTTTWithLN_57174604644763
// MI455X (gfx1250) — compile-verified
//
#include <hip/hip_runtime.h>
#include <cstdint>

typedef float v2f __attribute__((ext_vector_type(2)));
typedef float v8f __attribute__((ext_vector_type(8)));
typedef unsigned int uint32x4 __attribute__((ext_vector_type(4)));
typedef int int32x8 __attribute__((ext_vector_type(8)));
typedef int int32x4 __attribute__((ext_vector_type(4)));

#define LN_EPS 1e-5f

constexpr int Bb = 4, Ss = 1024, Dd = 1024, Hh = 16, dh = 64;
constexpr int BH = Bb * Hh;   // 64 heads total
constexpr int Mm = Bb * Ss;   // 4096 rows

// ---------------- WMMA fragment helpers (fp32, 16x16x4) ----------------
__device__ __forceinline__ v2f frag_a(const float* t, int m0, int k0, int ld) {
  int lane = threadIdx.x & 31;
  int m = lane & 15;
  int kk = (lane >> 4) << 1;           // 0 or 2
  const float* p = t + (m0 + m) * ld + k0 + kk;
  v2f r; r.x = p[0]; r.y = p[1];
  return r;
}
__device__ __forceinline__ v2f frag_b_kmajor(const float* t, int n0, int k0, int ld) {
  int lane = threadIdx.x & 31;
  int n = lane & 15;
  int kk = (lane >> 4) << 1;
  v2f r;
  r.x = t[(k0 + kk) * ld + n0 + n];
  r.y = t[(k0 + kk + 1) * ld + n0 + n];
  return r;
}
__device__ __forceinline__ v2f frag_b_nmajor(const float* t, int n0, int k0, int ld) {
  int lane = threadIdx.x & 31;
  int n = lane & 15;
  const float* p = t + (n0 + n) * ld + k0 + ((lane >> 4) << 1);
  v2f r; r.x = p[0]; r.y = p[1];
  return r;
}
__device__ __forceinline__ v8f wmma4(v2f a, v2f b, v8f c) {
  return __builtin_amdgcn_wmma_f32_16x16x4_f32(false, a, false, b, (short)0, c, false, false);
}
__device__ __forceinline__ v8f vzero8() { v8f z = {0.f,0.f,0.f,0.f,0.f,0.f,0.f,0.f}; return z; }

// ---------------- Tensor Data Mover: 2-D tile global -> LDS with row padding ----------------
// pad_interval_code: pad after (2<<code) DWORDs; pad_amount fixed at 1 DWORD
// (tile_w=16 + code 3 -> ld 17; tile_w=64 + code 5 -> ld 65)
__device__ __forceinline__ void tdm_load_tile_2d(
    const void* gptr, void* lptr,
    uint32_t tile_w, uint32_t tile_h,
    uint32_t tensor_w, uint32_t tensor_h,
    uint32_t row_stride_elems, uint32_t pad_interval_code) {
  uint64_t ga = (uint64_t)(uintptr_t)gptr;
  uint32_t lds = (uint32_t)(uintptr_t)lptr;   // LDS aperture: addr[31:0] is the LDS byte offset
  uint32x4 g0;
  g0[0] = 1u;                                   // count=1 valid user descriptor
  g0[1] = lds;                                  // lds_addr
  g0[2] = (uint32_t)ga;                         // global_addr[31:0]
  g0[3] = (uint32_t)((ga >> 32) & 0x01FFFFFFu)  // global_addr[56:32]
        | (2u << 30);                           // type = 2 ("image")
  int32x8 g1;
  g1[0] = (int)((2u << 16)                      // data_size = 4 bytes
        | (1u << 20)                            // pad_enable
        | (pad_interval_code << 22)             // pad interval
        | (0u << 25));                          // pad_amount = 1 DWORD
  g1[1] = (int)((tensor_w & 0xFFFFu) << 16);                       // tensor_dim0[15:0] @ bits63:48
  g1[2] = (int)((tensor_w >> 16) | ((tensor_h & 0xFFFFu) << 16));  // dim0 hi / dim1 lo
  g1[3] = (int)((tensor_h >> 16) | ((tile_w & 0xFFFFu) << 16));    // dim1 hi / tile_dim0
  g1[4] = (int)(tile_h & 0xFFFFu);                                 // tile_dim1 (tile_dim2 = 0)
  g1[5] = (int)row_stride_elems;                                   // tensor_dim0_stride[31:0]
  g1[6] = 0;
  g1[7] = 0;
  int32x4 z4 = {0, 0, 0, 0};
  int32x8 z8 = {0, 0, 0, 0, 0, 0, 0, 0};
  __builtin_amdgcn_tensor_load_to_lds(g0, g1, z4, z4, z8, 0);
}

// ---------------- Kernel 1: QKV projection GEMM (TDM double-buffered) ----------------
__global__ __launch_bounds__(128) void qkv_gemm(
    const float* __restrict__ x, const float* __restrict__ Wp, const float* __restrict__ bp,
    float* __restrict__ q, float* __restrict__ k, float* __restrict__ v) {
  __shared__ float As[2][64 * 17];
  __shared__ float Bs[2][16 * 65];
  const int tid = threadIdx.x, wave = tid >> 5, lane = tid & 31;
  const int m0 = blockIdx.x * 64;
  const int n0g = blockIdx.y * 64;
  const int which = n0g >> 10;
  const int h = (n0g & 1023) >> 6;
  float* dst = (which == 0) ? q : (which == 1) ? k : v;
  const int NK = Dd / 16;   // 64

  if (wave == 0) {
    tdm_load_tile_2d(x + (size_t)m0 * Dd, &As[0][0], 16, 64, Dd, Mm, Dd, 3);
    tdm_load_tile_2d(Wp + (size_t)n0g, &Bs[0][0], 64, 16, 3 * Dd, Dd, 3 * Dd, 5);
  }

  v8f acc[4];
#pragma unroll
  for (int i = 0; i < 4; i++) acc[i] = vzero8();

  for (int it = 0; it < NK; ++it) {
    const int cur = it & 1;
    if (wave == 0) {
      if (it + 1 < NK) {
        const int k0n = (it + 1) * 16;
        tdm_load_tile_2d(x + (size_t)m0 * Dd + k0n, &As[1 - cur][0], 16, 64, Dd, Mm, Dd, 3);
        tdm_load_tile_2d(Wp + (size_t)k0n * (3 * Dd) + n0g, &Bs[1 - cur][0], 64, 16, 3 * Dd, Dd, 3 * Dd, 5);
        __builtin_amdgcn_s_wait_tensorcnt((short)2);   // tile `it` complete (in-order)
      } else {
        __builtin_amdgcn_s_wait_tensorcnt((short)0);
      }
    }
    __syncthreads();
    const int ml = wave * 16;
#pragma unroll
    for (int kk = 0; kk < 16; kk += 4) {
      v2f a = frag_a(&As[cur][0], ml, kk, 17);
#pragma unroll
      for (int ct = 0; ct < 4; ct++) {
        v2f b = frag_b_kmajor(&Bs[cur][0], ct * 16, kk, 65);
        acc[ct] = wmma4(a, b, acc[ct]);
      }
    }
    __syncthreads();
  }
  const int hi = lane >> 4, nlo = lane & 15;
#pragma unroll
  for (int ct = 0; ct < 4; ct++) {
    int col = ct * 16 + nlo;
    float bias = bp[n0g + col];
#pragma unroll
    for (int r = 0; r < 8; r++) {
      int m = m0 + wave * 16 + r + 8 * hi;
      int b_ = m >> 10, s_ = m & 1023;
      dst[(((size_t)(b_ * Hh + h) * Ss) + s_) * dh + col] = acc[ct][r] + bias;
    }
  }
}

// ---------------- row LN-VJP helper: cotangent = ln(z) - v ----------------
__device__ __forceinline__ void ln_vjp_row(const float* zr, const float* vr,
                                           const float* lng, const float* lnb,
                                           float* dz_out) {
  float s1 = 0.f, s2 = 0.f;
  for (int j = 0; j < dh; j++) { float z = zr[j]; s1 += z; s2 += z * z; }
  float mu = s1 * (1.f / dh);
  float var = s2 * (1.f / dh) - mu * mu;
  float rstd = rsqrtf(var + LN_EPS);
  float mw = 0.f, mwx = 0.f;
  for (int j = 0; j < dh; j++) {
    float xh = (zr[j] - mu) * rstd;
    float g = lng[j];
    float w = (xh * g + lnb[j] - vr[j]) * g;
    mw += w; mwx += w * xh;
  }
  mw *= (1.f / dh); mwx *= (1.f / dh);
  for (int j = 0; j < dh; j++) {
    float xh = (zr[j] - mu) * rstd;
    float g = lng[j];
    float w = (xh * g + lnb[j] - vr[j]) * g;
    dz_out[j] = rstd * (w - mw - xh * mwx);
  }
}

// ---------------- Kernel 2: Z = k@W0, dz1 = LN-VJP(Z, y1-v) ----------------
__global__ __launch_bounds__(128) void ttt_z_dz1(
    const float* __restrict__ k, const float* __restrict__ v, const float* __restrict__ W0,
    const float* __restrict__ lng, const float* __restrict__ lnb,
    float* __restrict__ Z, float* __restrict__ dz1) {
  __shared__ float kt[64 * 65];
  __shared__ float w0s[64 * 65];
  __shared__ float zt[64 * 65];
  const int tid = threadIdx.x, wave = tid >> 5, lane = tid & 31;
  const int bh = blockIdx.y, t0 = blockIdx.x * 64;
  const float* kb = k + (size_t)bh * Ss * dh;

#pragma unroll 4
  for (int i = 0; i < 32; i++) {
    int e = tid + i * 128;
    int row = e >> 6, col = e & 63;
    kt[row * 65 + col] = kb[(size_t)(t0 + row) * dh + col];
    w0s[row * 65 + col] = W0[row * dh + col];
  }
  __syncthreads();

  const int ml = wave * 16;
  v8f acc[4];
#pragma unroll
  for (int i = 0; i < 4; i++) acc[i] = vzero8();
#pragma unroll
  for (int k0 = 0; k0 < dh; k0 += 4) {
    v2f a = frag_a(kt, ml, k0, 65);
#pragma unroll
    for (int ct = 0; ct < 4; ct++) {
      v2f b = frag_b_kmajor(w0s, ct * 16, k0, 65);
      acc[ct] = wmma4(a, b, acc[ct]);
    }
  }
  const int hi = lane >> 4, nlo = lane & 15;
#pragma unroll
  for (int ct = 0; ct < 4; ct++)
#pragma unroll
    for (int r = 0; r < 8; r++) {
      int rl = ml + r + 8 * hi, col = ct * 16 + nlo;
      float val = acc[ct][r];
      zt[rl * 65 + col] = val;
      Z[((size_t)bh * Ss + t0 + rl) * dh + col] = val;
    }
  __syncthreads();

  if (tid < 64) {
    int row = tid;
    float dz[dh];
    ln_vjp_row(&zt[row * 65], v + ((size_t)bh * Ss + t0 + row) * dh, lng, lnb, dz);
    for (int j = 0; j < dh; j++)
      dz1[((size_t)bh * Ss + t0 + row) * dh + j] = dz[j];
  }
}

// ---------------- Kernel 3: Z2 = Z - tril(k k^T) dz1 ; dzsum = dz1 + LN-VJP(Z2) ----------------
__global__ __launch_bounds__(128) void ttt_z2_dzsum(
    const float* __restrict__ k, const float* __restrict__ v,
    const float* __restrict__ Z, const float* __restrict__ dz1,
    const float* __restrict__ lng, const float* __restrict__ lnb,
    float* __restrict__ dzsum) {
  __shared__ float kt[64 * 65];
  __shared__ float sc[64 * 65];
  __shared__ float ks[2][64 * 65];
  __shared__ float ds[2][64 * 65];
  const int tid = threadIdx.x, wave = tid >> 5, lane = tid & 31;
  const int bh = blockIdx.y, t0 = blockIdx.x * 64;
  const float* kb = k + (size_t)bh * Ss * dh;
  const float* d1b = dz1 + (size_t)bh * Ss * dh;
  const int ml = wave * 16, hi = lane >> 4, nlo = lane & 15;

#pragma unroll 4
  for (int i = 0; i < 32; i++) {
    int e = tid + i * 128;
    int row = e >> 6, col = e & 63;
    kt[row * 65 + col] = kb[(size_t)(t0 + row) * dh + col];
  }
  if (wave == 0) {   // prefetch s-tile 0 via TDM
    tdm_load_tile_2d(kb, &ks[0][0], 64, 64, dh, Ss, dh, 5);
    tdm_load_tile_2d(d1b, &ds[0][0], 64, 64, dh, Ss, dh, 5);
  }
  v8f acc[4];
#pragma unroll
  for (int ct = 0; ct < 4; ct++)
#pragma unroll
    for (int r = 0; r < 8; r++)
      acc[ct][r] = Z[((size_t)bh * Ss + t0 + ml + r + 8 * hi) * dh + ct * 16 + nlo];
  __syncthreads();

  const int last = (int)blockIdx.x;
  for (int st = 0; st <= last; st++) {
    const int cur = st & 1;
    if (wave == 0) {
      if (st < last) {
        const int s0n = (st + 1) * 64;
        tdm_load_tile_2d(kb + (size_t)s0n * dh, &ks[1 - cur][0], 64, 64, dh, Ss, dh, 5);
        tdm_load_tile_2d(d1b + (size_t)s0n * dh, &ds[1 - cur][0], 64, 64, dh, Ss, dh, 5);
        __builtin_amdgcn_s_wait_tensorcnt((short)2);
      } else {
        __builtin_amdgcn_s_wait_tensorcnt((short)0);
      }
    }
    __syncthreads();
    const int s0 = st * 64;
    v8f kkf[4];
#pragma unroll
    for (int i = 0; i < 4; i++) kkf[i] = vzero8();
#pragma unroll
    for (int k0 = 0; k0 < dh; k0 += 4) {
      v2f a = frag_a(kt, ml, k0, 65);
#pragma unroll
      for (int ct = 0; ct < 4; ct++) {
        v2f b = frag_b_nmajor(&ks[cur][0], ct * 16, k0, 65);
        kkf[ct] = wmma4(a, b, kkf[ct]);
      }
    }
#pragma unroll
    for (int ct = 0; ct < 4; ct++)
#pragma unroll
      for (int r = 0; r < 8; r++) {
        int rl = ml + r + 8 * hi;
        int rowg = t0 + rl, colg = s0 + ct * 16 + nlo;
        sc[rl * 65 + ct * 16 + nlo] = (colg <= rowg) ? -kkf[ct][r] : 0.f;
      }
    __syncthreads();
#pragma unroll
    for (int k0 = 0; k0 < 64; k0 += 4) {
      v2f a = frag_a(sc, ml, k0, 65);
#pragma unroll
      for (int ct = 0; ct < 4; ct++) {
        v2f b = frag_b_kmajor(&ds[cur][0], ct * 16, k0, 65);
        acc[ct] = wmma4(a, b, acc[ct]);
      }
    }
    __syncthreads();
  }
#pragma unroll
  for (int ct = 0; ct < 4; ct++)
#pragma unroll
    for (int r = 0; r < 8; r++)
      sc[(ml + r + 8 * hi) * 65 + ct * 16 + nlo] = acc[ct][r];
  __syncthreads();
  if (tid < 64) {
    int row = tid;
    float dz[dh];
    ln_vjp_row(&sc[row * 65], v + ((size_t)bh * Ss + t0 + row) * dh, lng, lnb, dz);
    for (int j = 0; j < dh; j++) {
      size_t off = ((size_t)bh * Ss + t0 + row) * dh + j;
      dzsum[off] = dz1[off] + dz[j];
    }
  }
}

// ---------------- Kernel 4: ctx = q@W0 - tril(q k^T)(dz1+dz2) ----------------
__global__ __launch_bounds__(128) void ttt_ctx(
    const float* __restrict__ q, const float* __restrict__ k,
    const float* __restrict__ dzs, const float* __restrict__ W0,
    float* __restrict__ ctx) {
  __shared__ float qt[64 * 65];
  __shared__ float sc[64 * 65];
  __shared__ float ks[2][64 * 65];
  __shared__ float ds[2][64 * 65];
  const int tid = threadIdx.x, wave = tid >> 5, lane = tid & 31;
  const int bh = blockIdx.y, t0 = blockIdx.x * 64;
  const float* qb = q + (size_t)bh * Ss * dh;
  const float* kb = k + (size_t)bh * Ss * dh;
  const float* dzb = dzs + (size_t)bh * Ss * dh;
  const int ml = wave * 16, hi = lane >> 4, nlo = lane & 15;

#pragma unroll 4
  for (int i = 0; i < 32; i++) {
    int e = tid + i * 128;
    int row = e >> 6, col = e & 63;
    qt[row * 65 + col] = qb[(size_t)(t0 + row) * dh + col];
    sc[row * 65 + col] = W0[row * dh + col];   // borrow sc for W0 during init
  }
  if (wave == 0) {
    tdm_load_tile_2d(kb, &ks[0][0], 64, 64, dh, Ss, dh, 5);
    tdm_load_tile_2d(dzb, &ds[0][0], 64, 64, dh, Ss, dh, 5);
  }
  __syncthreads();
  v8f acc[4];
#pragma unroll
  for (int i = 0; i < 4; i++) acc[i] = vzero8();
#pragma unroll
  for (int k0 = 0; k0 < dh; k0 += 4) {
    v2f a = frag_a(qt, ml, k0, 65);
#pragma unroll
    for (int ct = 0; ct < 4; ct++) {
      v2f b = frag_b_kmajor(sc, ct * 16, k0, 65);
      acc[ct] = wmma4(a, b, acc[ct]);
    }
  }

  const int last = (int)blockIdx.x;
  for (int st = 0; st <= last; st++) {
    const int cur = st & 1;
    if (wave == 0) {
      if (st < last) {
        const int s0n = (st + 1) * 64;
        tdm_load_tile_2d(kb + (size_t)s0n * dh, &ks[1 - cur][0], 64, 64, dh, Ss, dh, 5);
        tdm_load_tile_2d(dzb + (size_t)s0n * dh, &ds[1 - cur][0], 64, 64, dh, Ss, dh, 5);
        __builtin_amdgcn_s_wait_tensorcnt((short)2);
      } else {
        __builtin_amdgcn_s_wait_tensorcnt((short)0);
      }
    }
    __syncthreads();   // also fences the init-GEMM reads of sc(W0) on st==0
    const int s0 = st * 64;
    v8f af[4];
#pragma unroll
    for (int i = 0; i < 4; i++) af[i] = vzero8();
#pragma unroll
    for (int k0 = 0; k0 < dh; k0 += 4) {
      v2f a = frag_a(qt, ml, k0, 65);
#pragma unroll
      for (int ct = 0; ct < 4; ct++) {
        v2f b = frag_b_nmajor(&ks[cur][0], ct * 16, k0, 65);
        af[ct] = wmma4(a, b, af[ct]);
      }
    }
#pragma unroll
    for (int ct = 0; ct < 4; ct++)
#pragma unroll
      for (int r = 0; r < 8; r++) {
        int rl = ml + r + 8 * hi;
        int rowg = t0 + rl, colg = s0 + ct * 16 + nlo;
        sc[rl * 65 + ct * 16 + nlo] = (colg <= rowg) ? -af[ct][r] : 0.f;
      }
    __syncthreads();
#pragma unroll
    for (int k0 = 0; k0 < 64; k0 += 4) {
      v2f a = frag_a(sc, ml, k0, 65);
#pragma unroll
      for (int ct = 0; ct < 4; ct++) {
        v2f b = frag_b_kmajor(&ds[cur][0], ct * 16, k0, 65);
        acc[ct] = wmma4(a, b, acc[ct]);
      }
    }
    __syncthreads();
  }
  const int b_ = bh >> 4, h = bh & 15;
#pragma unroll
  for (int ct = 0; ct < 4; ct++)
#pragma unroll
    for (int r = 0; r < 8; r++) {
      int rl = ml + r + 8 * hi, col = ct * 16 + nlo;
      ctx[((size_t)(b_ * Ss) + t0 + rl) * Dd + h * dh + col] = acc[ct][r];
    }
}

// ---------------- Kernel 5/6: deterministic group-norm stats ----------------
__global__ __launch_bounds__(256) void gn_partials(const float* __restrict__ ctx,
                                                   float* __restrict__ partial) {
  __shared__ float r1[256], r2[256];
  const int tid = threadIdx.x;
  const int bh = blockIdx.y, t0 = blockIdx.x * 64;
  const int b_ = bh >> 4, h = bh & 15;
  float s1 = 0.f, s2 = 0.f;
#pragma unroll 4
  for (int i = 0; i < 16; i++) {
    int e = tid + i * 256;
    int row = e >> 6, col = e & 63;
    float vx = ctx[((size_t)(b_ * Ss) + t0 + row) * Dd + h * dh + col];
    s1 += vx; s2 += vx * vx;
  }
  r1[tid] = s1; r2[tid] = s2;
  __syncthreads();
  for (int off = 128; off > 0; off >>= 1) {
    if (tid < off) { r1[tid] += r1[tid + off]; r2[tid] += r2[tid + off]; }
    __syncthreads();
  }
  if (tid == 0) {
    partial[(bh * 16 + blockIdx.x) * 2 + 0] = r1[0];
    partial[(bh * 16 + blockIdx.x) * 2 + 1] = r2[0];
  }
}
__global__ void gn_reduce(const float* __restrict__ partial, float* __restrict__ stats) {
  int bh = threadIdx.x;
  float s1 = 0.f, s2 = 0.f;
  for (int i = 0; i < 16; i++) {
    s1 += partial[(bh * 16 + i) * 2 + 0];
    s2 += partial[(bh * 16 + i) * 2 + 1];
  }
  stats[bh * 2 + 0] = s1;
  stats[bh * 2 + 1] = s2;
}

// ---------------- Kernel 7: out = groupnorm(ctx)@Wout + bout ----------------
__global__ __launch_bounds__(128) void out_gemm(
    const float* __restrict__ ctx, const float* __restrict__ stats,
    const float* __restrict__ gng, const float* __restrict__ gnb,
    const float* __restrict__ Wout, const float* __restrict__ bout,
    float* __restrict__ out) {
  __shared__ float As[64 * 17];
  __shared__ float Bs[2][16 * 65];
  const int tid = threadIdx.x, wave = tid >> 5, lane = tid & 31;
  const int m0 = blockIdx.x * 64;
  const int n0 = blockIdx.y * 64;
  const int b_ = m0 >> 10;
  const float inv = 1.f / (float)(Ss * dh);
  const int NK = Dd / 16;

  if (wave == 0)
    tdm_load_tile_2d(Wout + (size_t)n0, &Bs[0][0], 64, 16, Dd, Dd, Dd, 5);

  v8f acc[4];
#pragma unroll
  for (int i = 0; i < 4; i++) acc[i] = vzero8();

  for (int it = 0; it < NK; ++it) {
    const int cur = it & 1;
    const int k0g = it * 16;
    if (wave == 0 && it + 1 < NK)
      tdm_load_tile_2d(Wout + (size_t)(it + 1) * 16 * Dd + n0, &Bs[1 - cur][0], 64, 16, Dd, Dd, Dd, 5);
    // A tile with group-norm fused (overlaps the in-flight TDM)
#pragma unroll
    for (int r = 0; r < 8; r++) {
      int e = tid + r * 128;
      int row = e >> 4, kk = e & 15;
      int kc = k0g + kk;
      int h = kc >> 6;
      float sm = stats[(b_ * Hh + h) * 2 + 0] * inv;
      float sv = stats[(b_ * Hh + h) * 2 + 1] * inv - sm * sm;
      float rstd = rsqrtf(sv + LN_EPS);
      float cv = ctx[(size_t)(m0 + row) * Dd + kc];
      As[row * 17 + kk] = (cv - sm) * rstd * gng[kc] + gnb[kc];
    }
    if (wave == 0) {
      if (it + 1 < NK) __builtin_amdgcn_s_wait_tensorcnt((short)1);
      else             __builtin_amdgcn_s_wait_tensorcnt((short)0);
    }
    __syncthreads();
    const int ml = wave * 16;
#pragma unroll
    for (int kk = 0; kk < 16; kk += 4) {
      v2f a = frag_a(As, ml, kk, 17);
#pragma unroll
      for (int ct = 0; ct < 4; ct++) {
        v2f b = frag_b_kmajor(&Bs[cur][0], ct * 16, kk, 65);
        acc[ct] = wmma4(a, b, acc[ct]);
      }
    }
    __syncthreads();
  }
  const int hi = lane >> 4, nlo = lane & 15;
#pragma unroll
  for (int ct = 0; ct < 4; ct++) {
    int col = ct * 16 + nlo;
    float bias = bout[n0 + col];
#pragma unroll
    for (int r = 0; r < 8; r++) {
      int m = m0 + wave * 16 + r + 8 * hi;
      out[(size_t)m * Dd + n0 + col] = acc[ct][r] + bias;
    }
  }
}

// ---------------- host launch ----------------
extern "C" void kernel_launch(void* const* d_in, const int* in_sizes, int n_in,
                              void* d_out, int out_size, void* d_ws, size_t ws_size,
                              hipStream_t stream) {
  (void)in_sizes; (void)n_in; (void)out_size; (void)ws_size;
  const float* x     = (const float*)d_in[0];
  const float* Wproj = (const float*)d_in[1];
  const float* bproj = (const float*)d_in[2];
  const float* W0    = (const float*)d_in[3];
  const float* ln_g  = (const float*)d_in[4];
  const float* ln_b  = (const float*)d_in[5];
  const float* gn_g  = (const float*)d_in[6];
  const float* gn_b  = (const float*)d_in[7];
  const float* Wout  = (const float*)d_in[8];
  const float* bout  = (const float*)d_in[9];
  float* out = (float*)d_out;

  float* ws = (float*)d_ws;
  const size_t HS = (size_t)BH * Ss * dh;
  float* q    = ws + 0 * HS;
  float* k    = ws + 1 * HS;
  float* v    = ws + 2 * HS;
  float* Z    = ws + 3 * HS;
  float* dz1  = ws + 4 * HS;
  float* dzs  = ws + 5 * HS;
  float* ctx  = ws + 6 * HS;
  float* part = ws + 7 * HS;
  float* stat = part + 2048;

  qkv_gemm<<<dim3(Mm / 64, (3 * Dd) / 64), 128, 0, stream>>>(x, Wproj, bproj, q, k, v);
  ttt_z_dz1<<<dim3(Ss / 64, BH), 128, 0, stream>>>(k, v, W0, ln_g, ln_b, Z, dz1);
  ttt_z2_dzsum<<<dim3(Ss / 64, BH), 128, 0, stream>>>(k, v, Z, dz1, ln_g, ln_b, dzs);
  ttt_ctx<<<dim3(Ss / 64, BH), 128, 0, stream>>>(q, k, dzs, W0, ctx);
  gn_partials<<<dim3(Ss / 64, BH), 256, 0, stream>>>(ctx, part);
  gn_reduce<<<1, BH, 0, stream>>>(part, stat);
  out_gemm<<<dim3(Mm / 64, Dd / 64), 128, 0, stream>>>(ctx, stat, gn_g, gn_b, Wout, bout, out);
}